// SHConv_31585189494985
// MI455X (gfx1250) — compile-verified
//
#include <hip/hip_runtime.h>

// CDNA5 / gfx1250 SHConv: out[b,o,p] = sum_i W_{l(p)}[i,o] * x[b,i,p]
// fp32 WMMA 16x16x4 path (exact fp32 precision, ~memory-roofline bound).
//
// Per wave (one wave32 per block): fixed (degree l, 16-wide o-tile, batch b).
//  - A tile (16 out x 128 in) built once in registers by interpolating the
//    17 anchor matrices (weight is 1.1 MB -> L2-hot; redundancy is free).
//  - Loop over 16-wide p-tiles of degree l: stage the whole 128-deep B panel
//    into registers (one load clause), then run the 32-WMMA chain, then store.
//  - Next tile's x columns are prefetched (global_prefetch_b8) during compute.

typedef float v2f __attribute__((ext_vector_type(2)));
typedef float v8f __attribute__((ext_vector_type(8)));

#define IN_CH   128
#define OUT_CH  128
#define P_TOT   6561     // (L+1)^2, L=80
#define NCPT    17
#define NBATCH  16

__global__ __launch_bounds__(32)
void shconv_wmma_f32(const float* __restrict__ x,     // [16][128][6561]
                     const float* __restrict__ w,     // [128][128][17]
                     float* __restrict__ out)         // [16][128][6561]
{
    // grid: 81 degrees * 8 o-tiles * 16 batches, one wave32 per block
    const int blk  = blockIdx.x;
    const int b    = blk & 15;          // batch
    const int ot   = (blk >> 4) & 7;    // o-tile (16 out channels)
    const int l    = blk >> 7;          // SH degree 0..80
    const int lane = threadIdx.x;       // 0..31
    const int m    = lane & 15;         // M / N index within tile
    const int h    = lane >> 4;         // half-wave selector (K split)
    const int o0   = ot << 4;

    // ---- piecewise-linear interpolation parameters for degree l ----
    // l in [0,75): segment l/5, frac (l%5)/5 ; l in [75,80]: anchors 15,16
    int   c0; float t;
    if (l < 75) { c0 = l / 5;  t = (float)(l - 5 * c0) * 0.2f; }
    else        { c0 = 15;     t = (float)(l - 75)     * 0.2f; }
    const float u = 1.0f - t;

    // ---- build A tile in registers: rows M = o0+m, K = 0..127 ----
    // A VGPR layout (16x4 fp32): lane L, vgpr j -> K = 4s + j + 2*(L>>4),
    // so per k-step each lane holds the consecutive K pair (4s+2h, 4s+2h+1).
    v2f a[32];
    const int orow = o0 + m;
    #pragma unroll
    for (int s = 0; s < 32; ++s) {
        const int i0 = 4 * s + 2 * h;                       // K index (= in-ch)
        const float* wp0 = w + ((size_t)i0 * OUT_CH + orow) * NCPT + c0;
        const float* wp1 = wp0 + (size_t)OUT_CH * NCPT;     // in-ch i0+1
        a[s].x = u * wp0[0] + t * wp0[1];
        a[s].y = u * wp1[0] + t * wp1[1];
    }

    // ---- positions owned by degree l: [l*l, l*l + 2l] ----
    const int pbase  = l * l;
    const int np     = 2 * l + 1;
    const int ntiles = (np + 15) >> 4;
    const int pmax   = pbase + np - 1;

    const float* xb = x   + (size_t)b * IN_CH  * P_TOT;
    float*       ob = out + (size_t)b * OUT_CH * P_TOT;

    // per-lane row base for B loads: rows 2h and 2h+1, column varies
    const float* xrow = xb + (size_t)(2 * h) * P_TOT;

    for (int pt = 0; pt < ntiles; ++pt) {
        const int p  = pbase + (pt << 4) + m;       // this lane's column
        const int pc = (p > pmax) ? pmax : p;       // clamp loads (garbage
                                                    // columns never stored)

        // ---- stage full 128-deep B panel into registers: one load clause ----
        // B VGPR layout (4x16 fp32): lane L, vgpr j -> K = 4s + j + 2*(L>>4),
        // N = L&15  => lane loads x rows (4s+2h) and (4s+2h+1) at column pc.
        v2f bb[32];
        #pragma unroll
        for (int s = 0; s < 32; ++s) {
            bb[s].x = xrow[(size_t)(4 * s)     * P_TOT + pc];
            bb[s].y = xrow[(size_t)(4 * s + 1) * P_TOT + pc];
        }

        // ---- prefetch next p-tile's columns while the WMMA chain runs ----
        if (pt + 1 < ntiles) {
            int pn = pbase + ((pt + 1) << 4) + m;
            if (pn > pmax) pn = pmax;
            #pragma unroll
            for (int s = 0; s < 8; ++s) {   // one line-touch per 16 rows
                __builtin_prefetch(&xrow[(size_t)(16 * s) * P_TOT + pn], 0, 1);
            }
        }

        // ---- 32-step fp32 WMMA chain: D = A(16x4)*B(4x16) + C ----
        v8f c = {0.f, 0.f, 0.f, 0.f, 0.f, 0.f, 0.f, 0.f};
        #pragma unroll
        for (int s = 0; s < 32; ++s) {
            c = __builtin_amdgcn_wmma_f32_16x16x4_f32(
                    /*neg_a=*/false, a[s],
                    /*neg_b=*/false, bb[s],
                    /*c_mod=*/(short)0, c,
                    /*reuse_a=*/false, /*reuse_b=*/false);
        }

        // C/D layout: vgpr j -> out-row o0 + j + 8*h, column p (lane&15)
        if (p <= pmax) {
            #pragma unroll
            for (int j = 0; j < 8; ++j) {
                ob[(size_t)(o0 + j + 8 * h) * P_TOT + p] = c[j];
            }
        }
    }
}

extern "C" void kernel_launch(void* const* d_in, const int* in_sizes, int n_in,
                              void* d_out, int out_size, void* d_ws, size_t ws_size,
                              hipStream_t stream)
{
    (void)in_sizes; (void)n_in; (void)out_size; (void)d_ws; (void)ws_size;
    const float* x = (const float*)d_in[0];
    const float* w = (const float*)d_in[1];
    float*     out = (float*)d_out;

    // 81 degrees * 8 o-tiles * 16 batches = 10368 single-wave blocks
    shconv_wmma_f32<<<dim3(81 * 8 * 16), dim3(32), 0, stream>>>(x, w, out);
}